// AE_9775345566294
// MI455X (gfx1250) — compile-verified
//
#include <hip/hip_runtime.h>
#include <math.h>

// Problem constants (from reference setup_inputs)
#define B_N 50000
#define L_N 51
#define CR_N 3
#define H_N 5
#define G_N 15   // 3*H
#define F_N 10   // 2*H
#define BN_EPS 1e-3f

typedef __attribute__((ext_vector_type(2))) float v2f;
typedef __attribute__((ext_vector_type(8))) float v8f;

// __has_builtin for amdgcn builtins is only meaningful during the device pass;
// the host pass always reports false.
#if defined(__HIP_DEVICE_COMPILE__)
#if !__has_builtin(__builtin_amdgcn_wmma_f32_16x16x4_f32)
#error "wmma f32 16x16x4 builtin not available on this toolchain"
#endif
#endif

__device__ __forceinline__ float wave_reduce_add(float v) {
  // wave32 reduction
  for (int off = 16; off > 0; off >>= 1) v += __shfl_down(v, off, 32);
  return v;
}

__device__ __forceinline__ float sigmoidf_(float x) { return 1.f / (1.f + expf(-x)); }

// ---------------------------------------------------------------------------
// Kernel 1: cw2 = f(bits, noise); accumulate per-(l,channel) sum/sumsq over B
// ---------------------------------------------------------------------------
__global__ void k_cw2_stats(const float* __restrict__ ib, const float* __restrict__ fn,
                            const float* __restrict__ fb, const float* __restrict__ e,
                            const float* __restrict__ kk, float* __restrict__ cw2,
                            float* __restrict__ csum, float* __restrict__ csq) {
  int l = blockIdx.x;
  int b = blockIdx.y * blockDim.x + threadIdx.x;
  float p1 = 0.f, p2 = 0.f;
  if (b < B_N) {
    size_t i = (size_t)b * L_N + l;
    float bit = ib[i];
    float n1 = fn[i * 3] + fb[i * 3];
    float ind = (-(2.f * bit - 1.f) * n1 > 0.f) ? 1.f : 0.f;
    float p1l = e[0] * n1 * ind;
    float vh4, vh5;
    if (l == 0) { vh4 = 1.f; vh5 = -1.f; }
    else {
      size_t ip = i - 1;  // (b, l-1)
      float bp  = ib[ip];
      float n1p = fn[ip * 3];
      float n2p = fn[ip * 3 + 1] + fb[ip * 3 + 1];
      float n3p = fn[ip * 3 + 2] + fb[ip * 3 + 2];
      float targ = -kk[0] * n1p + kk[1] * n2p - kk[2] * n3p;
      vh4 = (bp == 0.f) ? tanhf(targ + kk[3]) :  1.f;
      vh5 = (bp == 1.f) ? tanhf(targ - kk[3]) : -1.f;
    }
    float corr = e[1] * (vh4 + vh5);
    p1 =  p1l - corr;
    p2 = -p1l - corr;
    cw2[i * 2]     = p1;
    cw2[i * 2 + 1] = p2;
  }
  float s1 = wave_reduce_add(p1);
  float s2 = wave_reduce_add(p2);
  float q1 = wave_reduce_add(p1 * p1);
  float q2 = wave_reduce_add(p2 * p2);
  if ((threadIdx.x & 31) == 0) {
    atomicAdd(&csum[l * 2],     s1);
    atomicAdd(&csum[l * 2 + 1], s2);
    atomicAdd(&csq [l * 2],     q1);
    atomicAdd(&csq [l * 2 + 1], q2);
  }
}

// ---------------------------------------------------------------------------
// Kernel 2: finalize mean/std (ddof=1) and per-(l,c) codeword coefficients
// ---------------------------------------------------------------------------
__global__ void k_finalize(const float* __restrict__ csum, const float* __restrict__ csq,
                           const float* __restrict__ wa, const float* __restrict__ wf4,
                           const float* __restrict__ wl5, float* __restrict__ cmean,
                           float* __restrict__ cstd, float* __restrict__ coef) {
  int t = threadIdx.x;
  for (int i = t; i < L_N * 2; i += blockDim.x) {
    float s = csum[i];
    float m = s / (float)B_N;
    float var = (csq[i] - s * s / (float)B_N) / (float)(B_N - 1);  // ddof=1
    cmean[i] = m;
    cstd[i]  = sqrtf(var);
  }
  float swa = 0.f, swf = 0.f, swl = 0.f;
  for (int c = 0; c < 3; c++) swa += wa[c] * wa[c];
  for (int c = 0; c < 4; c++) swf += wf4[c] * wf4[c];
  for (int c = 0; c < 5; c++) swl += wl5[c] * wl5[c];
  float rem = (float)(L_N - 4 - 5);
  float den = (rem + swf + swl) * swa;
  float scale = sqrtf((float)CR_N * (float)L_N / den);
  for (int i = t; i < L_N * 3; i += blockDim.x) {
    int l = i / 3, c = i % 3;
    float wt = 1.f;
    if (l < 4) wt = wf4[l];
    else if (l >= L_N - 5) wt = wl5[l - (L_N - 5)];
    coef[i] = scale * wa[c] * wt;
  }
}

// ---------------------------------------------------------------------------
// Kernel 3: codewords (output #1) and noisy = codewords + forward_noise
// ---------------------------------------------------------------------------
__global__ void k_codewords(const float* __restrict__ ib, const float* __restrict__ fn,
                            const float* __restrict__ cw2, const float* __restrict__ cmean,
                            const float* __restrict__ cstd, const float* __restrict__ coef,
                            float* __restrict__ out_cw, float* __restrict__ noisy) {
  size_t N = (size_t)B_N * L_N;
  size_t i = (size_t)blockIdx.x * blockDim.x + threadIdx.x;
  if (i >= N) return;
  int l = (int)(i % L_N);
  float phase = 2.f * ib[i] - 1.f;
  float z1 = (cw2[i * 2]     - cmean[l * 2])     / cstd[l * 2];
  float z2 = (cw2[i * 2 + 1] - cmean[l * 2 + 1]) / cstd[l * 2 + 1];
  float c0 = coef[l * 3]     * phase;
  float c1 = coef[l * 3 + 1] * z1;
  float c2 = coef[l * 3 + 2] * z2;
  out_cw[i * 3]     = c0;
  out_cw[i * 3 + 1] = c1;
  out_cw[i * 3 + 2] = c2;
  noisy[i * 3]     = c0 + fn[i * 3];
  noisy[i * 3 + 1] = c1 + fn[i * 3 + 1];
  noisy[i * 3 + 2] = c2 + fn[i * 3 + 2];
}

// ---------------------------------------------------------------------------
// GRU direction kernel. One wave = one 16-row batch tile. Recurrent matvec
// h(16x5) @ Whh^T(5x15) is done as two chained V_WMMA_F32_16X16X4_F32
// (K padded 5->8, N padded 15->16, C seeded with bhh broadcast). The D tile
// is spilled to a per-wave LDS tile; each lane gathers its r/z/n gate
// pre-activations and updates exactly the h components it holds in its
// A-matrix registers (no transpose needed).
//   A layout (16x4 f32): lanes 0-15 hold {K0,K1}, lanes 16-31 hold {K2,K3}
//   C/D layout: VGPR v, lane -> [M = v + 8*(lane>>4)][N = lane&15]
// ---------------------------------------------------------------------------
template <int K>
__global__ void k_gru(const float* __restrict__ in, const float* __restrict__ Wih,
                      const float* __restrict__ Whh, const float* __restrict__ bih,
                      const float* __restrict__ bhh, float* __restrict__ out,
                      int fo, int rev) {
  __shared__ float s_Wih[G_N * K];
  __shared__ float s_bih[G_N];
  __shared__ float s_tile[8][16 * 16];  // one 16x16 f32 D tile per wave

  for (int i = threadIdx.x; i < G_N * K; i += blockDim.x) s_Wih[i] = Wih[i];
  if (threadIdx.x < G_N) s_bih[threadIdx.x] = bih[threadIdx.x];
  __syncthreads();  // only barrier; before any wave exits

  const int ntiles = B_N / 16;  // 3125, exact
  int wv   = threadIdx.x >> 5;
  int lane = threadIdx.x & 31;
  int tile = blockIdx.x * (blockDim.x >> 5) + wv;
  if (tile >= ntiles) return;

  int half = lane >> 4;
  int n    = lane & 15;                 // B/C/D column; also this lane's batch row m
  size_t brow = (size_t)tile * 16 + n;  // global batch row
  bool nv = (n < G_N);

  // B matrices from Whh (row-major (15,5)): B[k][col] = Whh[col][k]
  v2f bm1, bm2;
  if (half == 0) {
    bm1.x = nv ? Whh[n * H_N + 0] : 0.f;  // K=0
    bm1.y = nv ? Whh[n * H_N + 1] : 0.f;  // K=1
    bm2.x = nv ? Whh[n * H_N + 4] : 0.f;  // K=4 (second WMMA, chunk K=0)
    bm2.y = 0.f;
  } else {
    bm1.x = nv ? Whh[n * H_N + 2] : 0.f;  // K=2
    bm1.y = nv ? Whh[n * H_N + 3] : 0.f;  // K=3
    bm2.x = 0.f;
    bm2.y = 0.f;
  }
  float bias = nv ? bhh[n] : 0.f;  // C[m][n] = bhh[n] broadcast
  v8f cbias;
#pragma unroll
  for (int v = 0; v < 8; v++) cbias[v] = bias;

  // Hidden state, held in A-matrix layout:
  //   half 0: a1 = {h0, h1}, a2 = {h4, 0};  half 1: a1 = {h2, h3}
  v2f a1 = {0.f, 0.f}, a2 = {0.f, 0.f};
  float* tp = &s_tile[wv][0];

  for (int t = 0; t < L_N; ++t) {
    int ta = rev ? (L_N - 1 - t) : t;

    // gh = h @ Whh^T + bhh : two chained f32 WMMAs (K=0..3, then K=4)
    v8f d = __builtin_amdgcn_wmma_f32_16x16x4_f32(false, a1, false, bm1,
                                                  (short)0, cbias, false, false);
    d     = __builtin_amdgcn_wmma_f32_16x16x4_f32(false, a2, false, bm2,
                                                  (short)0, d,     false, false);

    // Spill D tile to LDS: tp[m*16 + col] = gh[m][col]
#pragma unroll
    for (int v = 0; v < 8; v++) tp[(v + 8 * half) * 16 + n] = d[v];
    asm volatile("s_wait_dscnt 0x0" ::: "memory");  // stores visible before gathers

    // Per-row input projection operands
    float xr[K];
    const float* xp = in + ((size_t)brow * L_N + ta) * K;
#pragma unroll
    for (int k2 = 0; k2 < K; k2++) xr[k2] = xp[k2];

    float hold0 = a1.x, hold1 = a1.y, hold2 = a2.x;
    int nj = (half == 0) ? 3 : 2;  // half 0 owns j={0,1,4}, half 1 owns j={2,3}
    float hn_[3];
#pragma unroll
    for (int q = 0; q < 3; q++) {
      int j = (half == 0) ? ((q == 0) ? 0 : ((q == 1) ? 1 : 4))
                          : ((q == 0) ? 2 : 3);
      float ghr = tp[n * 16 + j];
      float ghz = tp[n * 16 + j + 5];
      float ghn = tp[n * 16 + j + 10];
      float ir  = s_bih[j];
      float iz  = s_bih[j + 5];
      float inn = s_bih[j + 10];
#pragma unroll
      for (int k2 = 0; k2 < K; k2++) {
        ir  += xr[k2] * s_Wih[j * K + k2];
        iz  += xr[k2] * s_Wih[(j + 5) * K + k2];
        inn += xr[k2] * s_Wih[(j + 10) * K + k2];
      }
      float r  = sigmoidf_(ir + ghr);
      float zz = sigmoidf_(iz + ghz);
      float ng = tanhf(inn + r * ghn);
      float hold = (q == 0) ? hold0 : ((q == 1) ? hold1 : hold2);
      float hnew = (1.f - zz) * ng + zz * hold;
      hn_[q] = hnew;
      if (q < nj) out[((size_t)brow * L_N + ta) * F_N + fo + j] = hnew;
    }
    a1.x = hn_[0];
    a1.y = hn_[1];
    a2.x = (half == 0) ? hn_[2] : 0.f;
  }
}

// ---------------------------------------------------------------------------
// BN stats over (B*L) rows x 10 features -> st[0..9]=sum, st[10..19]=sumsq
// ---------------------------------------------------------------------------
__global__ void k_bnstats(const float* __restrict__ g, float* __restrict__ st) {
  size_t N = (size_t)B_N * L_N;
  float acc[F_N] = {}, acc2[F_N] = {};
  for (size_t i = (size_t)blockIdx.x * blockDim.x + threadIdx.x; i < N;
       i += (size_t)gridDim.x * blockDim.x) {
#pragma unroll
    for (int f = 0; f < F_N; f++) {
      float v = g[i * F_N + f];
      acc[f] += v;
      acc2[f] += v * v;
    }
  }
#pragma unroll
  for (int f = 0; f < F_N; f++) {
    float s = wave_reduce_add(acc[f]);
    float q = wave_reduce_add(acc2[f]);
    if ((threadIdx.x & 31) == 0) {
      atomicAdd(&st[f], s);
      atomicAdd(&st[F_N + f], q);
    }
  }
}

// ---------------------------------------------------------------------------
// BN apply in place (ddof=0 variance, matching jnp.var)
// ---------------------------------------------------------------------------
__global__ void k_bnapply(float* __restrict__ g, const float* __restrict__ st,
                          const float* __restrict__ gamma, const float* __restrict__ beta) {
  size_t N = (size_t)B_N * L_N * F_N;
  size_t i = (size_t)blockIdx.x * blockDim.x + threadIdx.x;
  if (i >= N) return;
  int f = (int)(i % F_N);
  float Nf = (float)((size_t)B_N * L_N);
  float m = st[f] / Nf;
  float var = st[F_N + f] / Nf - m * m;
  g[i] = gamma[f] * (g[i] - m) * rsqrtf(var + BN_EPS) + beta[f];
}

// ---------------------------------------------------------------------------
// Final: BN2 + linear(10->1) + sigmoid fused (output #2)
// ---------------------------------------------------------------------------
__global__ void k_final(const float* __restrict__ g2, const float* __restrict__ st,
                        const float* __restrict__ gamma, const float* __restrict__ beta,
                        const float* __restrict__ W, const float* __restrict__ bsc,
                        float* __restrict__ out) {
  size_t N = (size_t)B_N * L_N;
  size_t i = (size_t)blockIdx.x * blockDim.x + threadIdx.x;
  if (i >= N) return;
  float Nf = (float)N;
  float acc = bsc[0];
#pragma unroll
  for (int f = 0; f < F_N; f++) {
    float m = st[f] / Nf;
    float var = st[F_N + f] / Nf - m * m;
    float v = gamma[f] * (g2[i * F_N + f] - m) * rsqrtf(var + BN_EPS) + beta[f];
    acc += W[f] * v;
  }
  out[i] = 1.f / (1.f + expf(-acc));
}

// ---------------------------------------------------------------------------
// Launch. Workspace needs 25*B*L + 601 floats (~255 MB):
//   cw2 (2*NBL) | noisy (3*NBL) | g1 (10*NBL) | g2 (10*NBL) | stats (601)
// ---------------------------------------------------------------------------
extern "C" void kernel_launch(void* const* d_in, const int* in_sizes, int n_in,
                              void* d_out, int out_size, void* d_ws, size_t ws_size,
                              hipStream_t stream) {
  (void)in_sizes; (void)n_in; (void)out_size; (void)ws_size;
  const float* ib  = (const float*)d_in[0];
  const float* fn  = (const float*)d_in[1];
  const float* fb  = (const float*)d_in[2];
  const float* e   = (const float*)d_in[3];
  const float* kk  = (const float*)d_in[4];
  const float* wa  = (const float*)d_in[5];
  const float* wf4 = (const float*)d_in[6];
  const float* wl5 = (const float*)d_in[7];
  float* outp = (float*)d_out;

  const size_t NBL = (size_t)B_N * L_N;
  float* ws    = (float*)d_ws;
  float* cw2   = ws;
  float* noisy = cw2 + 2 * NBL;
  float* g1    = noisy + 3 * NBL;
  float* g2    = g1 + 10 * NBL;
  float* st    = g2 + 10 * NBL;
  float* csum  = st;          // L*2
  float* csq   = st + 102;    // L*2
  float* cmean = st + 204;    // L*2
  float* cstd  = st + 306;    // L*2
  float* coef  = st + 408;    // L*3
  float* bn1   = st + 561;    // 20
  float* bn2   = st + 581;    // 20

  (void)hipMemsetAsync(st, 0, 601 * sizeof(float), stream);

  dim3 grd1(L_N, (B_N + 255) / 256);
  k_cw2_stats<<<grd1, 256, 0, stream>>>(ib, fn, fb, e, kk, cw2, csum, csq);
  k_finalize<<<1, 256, 0, stream>>>(csum, csq, wa, wf4, wl5, cmean, cstd, coef);

  int nb = (int)((NBL + 255) / 256);
  k_codewords<<<nb, 256, 0, stream>>>(ib, fn, cw2, cmean, cstd, coef, outp, noisy);

  int gruBlocks = (B_N / 16 + 7) / 8;  // 8 waves (tiles) per 256-thread block
  k_gru<3><<<gruBlocks, 256, 0, stream>>>(noisy, (const float*)d_in[8],
      (const float*)d_in[9], (const float*)d_in[10], (const float*)d_in[11], g1, 0, 0);
  k_gru<3><<<gruBlocks, 256, 0, stream>>>(noisy, (const float*)d_in[12],
      (const float*)d_in[13], (const float*)d_in[14], (const float*)d_in[15], g1, 5, 1);

  k_bnstats<<<512, 256, 0, stream>>>(g1, bn1);
  int nb10 = (int)((NBL * 10 + 255) / 256);
  k_bnapply<<<nb10, 256, 0, stream>>>(g1, bn1, (const float*)d_in[24],
                                      (const float*)d_in[25]);

  k_gru<10><<<gruBlocks, 256, 0, stream>>>(g1, (const float*)d_in[16],
      (const float*)d_in[17], (const float*)d_in[18], (const float*)d_in[19], g2, 0, 0);
  k_gru<10><<<gruBlocks, 256, 0, stream>>>(g1, (const float*)d_in[20],
      (const float*)d_in[21], (const float*)d_in[22], (const float*)d_in[23], g2, 5, 1);

  k_bnstats<<<512, 256, 0, stream>>>(g2, bn2);
  k_final<<<nb, 256, 0, stream>>>(g2, bn2, (const float*)d_in[26],
      (const float*)d_in[27], (const float*)d_in[28], (const float*)d_in[29],
      outp + 3 * NBL);
}